// RNNCBNET_22582938042920
// MI455X (gfx1250) — compile-verified
//
#include <hip/hip_runtime.h>
#include <math.h>

#define NODE  400
#define TRS   40
#define STEPS 10
#define OUT   64
#define DMAX  500
#define RING  512
#define DTF   0.1f

// ---- workspace layout (float offsets) ----
#define STATS_OFF 0                          // 8 floats: [0]=ssq_b [1]=ssq_f [2]=ssq_l
#define RSL_OFF   8
#define RSF_OFF   (RSL_OFF + NODE)
#define RSB_OFF   (RSF_OFF + NODE)
#define LMDT_OFF  (RSB_OFF + NODE)           // NODE*OUT, K-major lm_d^T
#define VP_OFF    (LMDT_OFF + NODE*OUT)      // 48*NODE (M padded 40->48, zeroed)
#define WDL_OFF   (VP_OFF + 48*NODE)         // NODE*RING
#define WDF_OFF   (WDL_OFF + NODE*RING)
#define WDB_OFF   (WDF_OFF + NODE*RING)

typedef __attribute__((ext_vector_type(2))) float v2f;
typedef __attribute__((ext_vector_type(8))) float v8f;

__device__ __forceinline__ float relu_(float x){ return x > 0.f ? x : 0.f; }

// ---------------- 1) zero stats + padded Vp ----------------
__global__ void init_ws(float* ws){
    int i = blockIdx.x * blockDim.x + threadIdx.x;
    if (i < 8) ws[STATS_OFF + i] = 0.f;
    if (i < 48*NODE) ws[VP_OFF + i] = 0.f;
}

// ---------------- 2) Frobenius sums of squares ----------------
__global__ void sumsq_kernel(const float* __restrict__ wbb, const float* __restrict__ wff,
                             const float* __restrict__ wll, const float* __restrict__ sc,
                             float* ws){
    int j = blockIdx.x * blockDim.x + threadIdx.x;
    float pb = 0.f, pf = 0.f, pl = 0.f;
    if (j < NODE*NODE){
        int r = j / NODE, c = j - r*NODE;
        int jt = c*NODE + r;
        float s  = sc[j];
        float wb = __expf(wbb[j]) * s;
        float wf = __expf(wff[j]) * s;
        float wl = 0.5f * (__expf(wll[j]) * s + __expf(wll[jt]) * sc[jt]);
        pb = wb*wb; pf = wf*wf; pl = wl*wl;
    }
    #pragma unroll
    for (int m = 16; m; m >>= 1){
        pb += __shfl_xor(pb, m, 32);
        pf += __shfl_xor(pf, m, 32);
        pl += __shfl_xor(pl, m, 32);
    }
    __shared__ float sb[8], sf[8], sl[8];
    int wv = threadIdx.x >> 5, ln = threadIdx.x & 31;
    if (ln == 0){ sb[wv] = pb; sf[wv] = pf; sl[wv] = pl; }
    __syncthreads();
    if (threadIdx.x == 0){
        float tb = 0.f, tf = 0.f, tl = 0.f;
        for (int w = 0; w < 8; ++w){ tb += sb[w]; tf += sf[w]; tl += sl[w]; }
        atomicAdd(&ws[STATS_OFF+0], tb);
        atomicAdd(&ws[STATS_OFF+1], tf);
        atomicAdd(&ws[STATS_OFF+2], tl);
    }
}

// ---------------- 3) delay-binned weights Wd[i,d] + row sums ----------------
__global__ __launch_bounds__(256) void build_wd(const float* __restrict__ wbb, const float* __restrict__ wff,
                         const float* __restrict__ wll, const float* __restrict__ sc,
                         const float* __restrict__ dist, const float* __restrict__ theta,
                         float* ws){
    __shared__ float lL[RING], lF[RING], lB[RING];
    int i = blockIdx.x, tid = threadIdx.x;
    for (int d = tid; d < RING; d += 256){ lL[d] = 0.f; lF[d] = 0.f; lB[d] = 0.f; }
    __syncthreads();

    float inv_mu = 1.0f / (0.1f + relu_(theta[20]));
    float invnb = rsqrtf(ws[STATS_OFF+0]);
    float invnf = rsqrtf(ws[STATS_OFF+1]);
    float invnl = rsqrtf(ws[STATS_OFF+2]);

    for (int j = tid; j < NODE; j += 256){
        int idx = i*NODE + j, idxT = j*NODE + i;
        float s  = sc[idx];
        float wb = __expf(wbb[idx]) * s * invnb;
        float wf = __expf(wff[idx]) * s * invnf;
        float wl = 0.5f * (__expf(wll[idx]) * s + __expf(wll[idxT]) * sc[idxT]) * invnl;
        int d = (int)(dist[idx] * inv_mu);           // trunc toward zero (dist >= 0)
        d = d < 0 ? 0 : (d > DMAX-1 ? DMAX-1 : d);
        atomicAdd(&lL[d], wl);                        // ds_add_f32
        atomicAdd(&lF[d], wf);
        atomicAdd(&lB[d], wb);
    }
    __syncthreads();

    for (int d = tid; d < RING; d += 256){
        ws[WDL_OFF + i*RING + d] = lL[d];
        ws[WDF_OFF + i*RING + d] = lF[d];
        ws[WDB_OFF + i*RING + d] = lB[d];
    }
    // row sums rs = sum_d Wd[i,d]
    float pl = lL[tid] + lL[tid+256];
    float pf = lF[tid] + lF[tid+256];
    float pb = lB[tid] + lB[tid+256];
    #pragma unroll
    for (int m = 16; m; m >>= 1){
        pl += __shfl_xor(pl, m, 32);
        pf += __shfl_xor(pf, m, 32);
        pb += __shfl_xor(pb, m, 32);
    }
    __shared__ float rl[8], rf[8], rb[8];
    int wv = tid >> 5, ln = tid & 31;
    if (ln == 0){ rl[wv] = pl; rf[wv] = pf; rb[wv] = pb; }
    __syncthreads();
    if (tid == 0){
        float a = 0.f, b = 0.f, c = 0.f;
        for (int w = 0; w < 8; ++w){ a += rl[w]; b += rf[w]; c += rb[w]; }
        ws[RSL_OFF + i] = a; ws[RSF_OFF + i] = b; ws[RSB_OFF + i] = c;
    }
}

// ---------------- 4) lm_d^T (mean-centered, K-major) ----------------
__global__ void build_lmdt(const float* __restrict__ lm, float* ws){
    int k = blockIdx.x * blockDim.x + threadIdx.x;
    if (k >= NODE) return;
    float sum = 0.f;
    for (int o = 0; o < OUT; ++o) sum += lm[o*NODE + k];
    float mean = sum * (1.0f / OUT);
    for (int o = 0; o < OUT; ++o) ws[LMDT_OFF + k*OUT + o] = lm[o*NODE + k] - mean;
}

// ---------------- 5) per-node sim: one wave32 per node ----------------
__global__ __launch_bounds__(256) void sim_kernel(const float* __restrict__ external,
                                                  const float* __restrict__ hx,
                                                  const float* __restrict__ hE,
                                                  const float* __restrict__ theta,
                                                  float* ws){
    __shared__ float ring[8 * RING];
    int tid = threadIdx.x, wv = tid >> 5, lane = tid & 31;
    int node = blockIdx.x * 8 + wv;
    int wbase = wv * RING;

    // theta-derived constants (replicated per lane; scalar-unit friendly)
    float VL = relu_(theta[0]),  VI = relu_(theta[1]),  VE = relu_(theta[2]),  VN = relu_(theta[3]);
    float amg = relu_(theta[4]), VR = relu_(theta[5]),  psig = relu_(theta[6]);
    float gL = relu_(theta[7]),  invC = 1.f / relu_(theta[8]), kap = relu_(theta[9]);
    float gEg = relu_(theta[10]), gEsc = relu_(theta[11]);
    float gIg = relu_(theta[12]), gIsc = relu_(theta[13]);
    float gNg = relu_(theta[14]), gNsc = relu_(theta[15]);
    float gK  = relu_(theta[16]);
    float kki = relu_(theta[21]) * theta[23];
    float g_l = relu_(theta[24]), g_f = relu_(theta[25]), g_b = relu_(theta[26]);

    // delay weights into registers: lane owns d = t*32 + lane (conflict-free LDS pattern)
    float wl[16], wf[16], wb[16];
    const float* WL = &ws[WDL_OFF + node*RING];
    const float* WF = &ws[WDF_OFF + node*RING];
    const float* WB = &ws[WDB_OFF + node*RING];
    #pragma unroll
    for (int t = 0; t < 16; ++t){
        int d = t*32 + lane;
        wl[t] = WL[d]; wf[t] = WF[d]; wb[t] = WB[d];
    }
    float rsl = ws[RSL_OFF + node], rsf = ws[RSF_OFF + node], rsb = ws[RSB_OFF + node];

    // history ring: ring[(h - d) & 511] = sP from d+1 steps ago; init h = 0
    #pragma unroll
    for (int t = 0; t < 16; ++t){
        int d = t*32 + lane;
        ring[wbase + ((-d) & (RING-1))] = (d < DMAX) ? hE[node*DMAX + d] : 0.f;
    }
    int h = 0;

    // ODE state per population
    float V0 = hx[node*12+0], gEa = hx[node*12+1], gIa = hx[node*12+2],  gNa = hx[node*12+3];
    float V1 = hx[node*12+4], gEb = hx[node*12+5], gIb = hx[node*12+6],  gNb = hx[node*12+7];
    float V2 = hx[node*12+8], gEc = hx[node*12+9], gIc = hx[node*12+10], gNc = hx[node*12+11];
    __syncthreads();

    for (int tr = 0; tr < TRS; ++tr){
        for (int st = 0; st < STEPS; ++st){
            float u = kki * external[node*(STEPS*TRS) + st*TRS + tr];
            float sE = 1.f / (1.f + __expf(-psig * (V0 - VR)));
            float sI = 1.f / (1.f + __expf(-psig * (V1 - VR)));
            float sP = 1.f / (1.f + __expf(-psig * (V2 - VR)));

            // three simultaneous delay dots over the shared ring
            float al = 0.f, af = 0.f, ab = 0.f;
            #pragma unroll
            for (int t = 0; t < 16; ++t){
                int d = t*32 + lane;
                float v = ring[wbase + ((h - d) & (RING-1))];
                al = fmaf(wl[t], v, al);
                af = fmaf(wf[t], v, af);
                ab = fmaf(wb[t], v, ab);
            }
            #pragma unroll
            for (int m = 16; m; m >>= 1){
                al += __shfl_xor(al, m, 32);
                af += __shfl_xor(af, m, 32);
                ab += __shfl_xor(ab, m, 32);
            }
            float lrl = g_l * (al - rsl*sP);
            float lrf = g_f * (af - rsf*sP);
            float lrb = g_b * (ab - rsb*sP);

            float exc0 = gEg*sP + lrl + u, exc1 = gEsc*sE + lrb, exc2 = gK*sE + lrf + u;
            float inh0 = gIg*sI,           inh1 = gIsc*sI,       inh2 = gIg*sI;
            float nm0  = gNg*sP + lrl,     nm1  = gNsc*sE,       nm2  = gNg*sE;

            float mN0 = 1.f / (1.f + 0.2f * __expf(-amg * V0));
            float mN1 = 1.f / (1.f + 0.2f * __expf(-amg * V1));
            float mN2 = 1.f / (1.f + 0.2f * __expf(-amg * V2));

            float dV0 = (gL*(-VL - V0) + gEa*(VE - V0) + gIa*(-VI - V0) + gNa*mN0*(VN - V0)) * invC;
            float dV1 = (gL*(-VL - V1) + gEb*(VE - V1) + gIb*(-VI - V1) + gNb*mN1*(VN - V1)) * invC;
            float dV2 = (gL*(-VL - V2) + gEc*(VE - V2) + gIc*(-VI - V2) + gNc*mN2*(VN - V2)) * invC;

            V0 += DTF*dV0; V1 += DTF*dV1; V2 += DTF*dV2;
            gEa += DTF*kap*(exc0 - gEa); gEb += DTF*kap*(exc1 - gEb); gEc += DTF*kap*(exc2 - gEc);
            gIa += DTF*kap*(inh0 - gIa); gIb += DTF*kap*(inh1 - gIb); gIc += DTF*kap*(inh2 - gIc);
            gNa += DTF*kap*(nm0 - gNa);  gNb += DTF*kap*(nm1 - gNb);  gNc += DTF*kap*(nm2 - gNc);

            // push this step's sP (computed from pre-update V) into the ring
            h = (h + 1) & (RING-1);
            if (lane == 0) ring[wbase + h] = sP;
        }
        if (lane == 0) ws[VP_OFF + tr*NODE + node] = V2;   // pyramidal V after trial
    }
}

// ---------------- 6) EEG readout: (48x400) @ (400x64) via V_WMMA_F32_16X16X4_F32 ----------------
__global__ __launch_bounds__(32) void eeg_wmma(const float* __restrict__ Vp,
                                               const float* __restrict__ lmdT,
                                               const float* __restrict__ theta,
                                               float* __restrict__ out){
    int lane = threadIdx.x;
    int mt = blockIdx.x >> 2, nt = blockIdx.x & 3;      // 3 x 4 tiles
    float cy0 = theta[22], y0 = theta[19];

    int m    = mt*16 + (lane & 15);
    int koff = (lane >> 4) * 2;                          // lanes 0-15: K0/K1, lanes 16-31: K2/K3
    int n    = nt*16 + (lane & 15);

    v8f acc = {};
    for (int kc = 0; kc < 100; ++kc){
        int k0 = kc*4 + koff;
        v2f a, b;
        a.x = Vp[m*NODE + k0];
        a.y = Vp[m*NODE + k0 + 1];
        b.x = lmdT[k0*OUT + n];
        b.y = lmdT[(k0+1)*OUT + n];
        acc = __builtin_amdgcn_wmma_f32_16x16x4_f32(false, a, false, b,
                                                    (short)0, acc, false, false);
    }
    #pragma unroll
    for (int r = 0; r < 8; ++r){
        int mm = mt*16 + r + ((lane >> 4) << 3);
        int nn = nt*16 + (lane & 15);
        if (mm < TRS) out[mm*OUT + nn] = cy0*acc[r] - y0;
    }
}

// ---------------- launcher ----------------
extern "C" void kernel_launch(void* const* d_in, const int* in_sizes, int n_in,
                              void* d_out, int out_size, void* d_ws, size_t ws_size,
                              hipStream_t stream){
    const float* external = (const float*)d_in[0];
    const float* hx       = (const float*)d_in[1];
    const float* hE       = (const float*)d_in[2];
    const float* sc       = (const float*)d_in[3];
    const float* dist     = (const float*)d_in[4];
    const float* wbb      = (const float*)d_in[5];
    const float* wff      = (const float*)d_in[6];
    const float* wll      = (const float*)d_in[7];
    const float* lm       = (const float*)d_in[8];
    const float* theta    = (const float*)d_in[9];
    float* ws  = (float*)d_ws;
    float* out = (float*)d_out;

    init_ws    <<<(48*NODE + 255)/256, 256, 0, stream>>>(ws);
    sumsq_kernel<<<(NODE*NODE + 255)/256, 256, 0, stream>>>(wbb, wff, wll, sc, ws);
    build_wd   <<<NODE, 256, 0, stream>>>(wbb, wff, wll, sc, dist, theta, ws);
    build_lmdt <<<(NODE + 255)/256, 256, 0, stream>>>(lm, ws);
    sim_kernel <<<NODE/8, 256, 0, stream>>>(external, hx, hE, theta, ws);
    eeg_wmma   <<<12, 32, 0, stream>>>(ws + VP_OFF, ws + LMDT_OFF, theta, out);
}